// Net_32083405701629
// MI455X (gfx1250) — compile-verified
//
#include <hip/hip_runtime.h>
#include <math.h>

typedef float v2f __attribute__((ext_vector_type(2)));
typedef float v8f __attribute__((ext_vector_type(8)));

#define D 128
#define PB 64   // pooling partial blocks

struct SelState { unsigned prefix, krem, total_gt, pad; };

// ---------------- edges ----------------
__global__ void k_edges_init(const int* __restrict__ ei, int* es, int* ed, int* em, int E){
  int i = blockIdx.x*blockDim.x + threadIdx.x;
  if (i < E){ es[i]=ei[i]; ed[i]=ei[E+i]; em[i]=1; }
}

// ---------------- layer 0: h is (n,1), outer products ----------------
__global__ void k_layer0(const float* __restrict__ x, const float* __restrict__ W1,
                         const float* __restrict__ V1, const float* __restrict__ b0,
                         float* hw, float* acc, int n){
  int i = blockIdx.x*blockDim.x + threadIdx.x;
  if (i < n*D){ int r=i>>7, c=i&127; float xv = x[r];
    hw[i] = xv*W1[c]; acc[i] = xv*V1[c] + b0[c]; }
}

// ---------------- fp32 WMMA GEMM: C[n,128] = A[n,128] @ B[128,128] (+bias) ----------------
// Wave computes a 64x16 tile (4 accumulators). Out-of-range rows are clamped to
// row 0 WITHOUT masking: in WMMA, D-row M depends only on A-row M, and the
// store loop guards m < n, so garbage rows are never written. K loop is fully
// branch-free: A = global_load_b64, B = 2x global_load_b32, 4 WMMAs per step.
__global__ void k_gemm(const float* __restrict__ A, const float* __restrict__ B,
                       const float* __restrict__ bias, float* __restrict__ C, int n){
  int wid  = threadIdx.x >> 5;          // 8 waves -> 8 column tiles
  int lane = threadIdx.x & 31;
  int half = lane >> 4, l = lane & 15;
  int n0 = wid * 16;
  int m0 = blockIdx.x * 64;

  int r0 = m0      + l, r1 = m0 + 16 + l, r2 = m0 + 32 + l, r3 = m0 + 48 + l;
  const float* a0 = A + (size_t)(r0 < n ? r0 : 0) * D + 2*half;
  const float* a1 = A + (size_t)(r1 < n ? r1 : 0) * D + 2*half;
  const float* a2 = A + (size_t)(r2 < n ? r2 : 0) * D + 2*half;
  const float* a3 = A + (size_t)(r3 < n ? r3 : 0) * D + 2*half;

  v8f c0 = {0.f,0.f,0.f,0.f,0.f,0.f,0.f,0.f};
  v8f c1 = c0, c2 = c0, c3 = c0;

  #pragma unroll 8
  for (int k0 = 0; k0 < D; k0 += 4){
    int kb = k0 + 2*half;
    v2f b;
    b[0] = B[(size_t)kb*D     + n0 + l];
    b[1] = B[(size_t)(kb+1)*D + n0 + l];

    v2f av0 = *(const v2f*)(a0 + k0);
    v2f av1 = *(const v2f*)(a1 + k0);
    v2f av2 = *(const v2f*)(a2 + k0);
    v2f av3 = *(const v2f*)(a3 + k0);

    c0 = __builtin_amdgcn_wmma_f32_16x16x4_f32(false, av0, false, b, (short)0, c0, false, false);
    c1 = __builtin_amdgcn_wmma_f32_16x16x4_f32(false, av1, false, b, (short)0, c1, false, false);
    c2 = __builtin_amdgcn_wmma_f32_16x16x4_f32(false, av2, false, b, (short)0, c2, false, false);
    c3 = __builtin_amdgcn_wmma_f32_16x16x4_f32(false, av3, false, b, (short)0, c3, false, false);
  }

  float bv = bias ? bias[n0 + l] : 0.f;
  #pragma unroll
  for (int r = 0; r < 8; ++r){
    int mr = r + 8*half;
    int m = m0 + mr;
    if (m      < n) C[(size_t)(m     )*D + n0 + l] = c0[r] + bv;
    if (m + 16 < n) C[(size_t)(m + 16)*D + n0 + l] = c1[r] + bv;
    if (m + 32 < n) C[(size_t)(m + 32)*D + n0 + l] = c2[r] + bv;
    if (m + 48 < n) C[(size_t)(m + 48)*D + n0 + l] = c3[r] + bv;
  }
}

// ---------------- degree / norm ----------------
__global__ void k_zero(float* p, int n){
  int i = blockIdx.x*blockDim.x + threadIdx.x; if (i<n) p[i]=0.f;
}
__global__ void k_edge_deg(const int* __restrict__ ed, const int* __restrict__ em,
                           float* deg, int E){
  int i = blockIdx.x*blockDim.x + threadIdx.x;
  if (i<E && em[i]) atomicAdd(&deg[ed[i]], 1.0f);
}
__global__ void k_dinv(const float* __restrict__ deg, float* dinv, int n){
  int i = blockIdx.x*blockDim.x + threadIdx.x;
  if (i<n){ float d = deg[i]; dinv[i] = d>0.f ? rsqrtf(d) : 0.f; }
}

// ---------------- edge aggregation: acc[dst] += norm * hw[src] ----------------
__global__ void k_edge_agg(const int* __restrict__ es, const int* __restrict__ ed,
                           const int* __restrict__ em, const float* __restrict__ dinv,
                           const float* __restrict__ hw, float* acc, int E){
  int gid = blockIdx.x*blockDim.x + threadIdx.x;
  int e = gid >> 5;
  int c = (gid & 31) << 2;
  if (e < E && em[e]){
    int s = es[e], d = ed[e];
    float w = dinv[s]*dinv[d];
    const float4 v = *(const float4*)(hw + (size_t)s*D + c);
    float* p = acc + (size_t)d*D + c;
    atomicAdd(p+0, v.x*w); atomicAdd(p+1, v.y*w);
    atomicAdd(p+2, v.z*w); atomicAdd(p+3, v.w*w);
  }
}

// ---------------- relu + BN(eval) + PReLU ----------------
__global__ void k_post(const float* __restrict__ acc, const float* __restrict__ g,
                       const float* __restrict__ be, const float* __restrict__ mu,
                       const float* __restrict__ va, const float* __restrict__ pa,
                       float* h, int n){
  int i = blockIdx.x*blockDim.x + threadIdx.x;
  if (i < n*D){
    int c = i & 127;
    float v = fmaxf(acc[i], 0.f);
    v = (v - mu[c]) * (g[c] * rsqrtf(va[c] + 1e-5f)) + be[c];
    float a = pa[0];
    h[i] = v > 0.f ? v : a*v;
  }
}

// ---------------- topk scores: tanh(h.p/||p||), sortable keys ----------------
__global__ void k_score(const float* __restrict__ h, const float* __restrict__ p,
                        float* score, unsigned* keys, int n){
  int wid = threadIdx.x >> 5, lane = threadIdx.x & 31;
  int row = blockIdx.x*8 + wid;
  if (row >= n) return;
  float dot = 0.f, pp = 0.f;
  int c0 = lane*4;
  #pragma unroll
  for (int j=0;j<4;++j){ float pv = p[c0+j]; dot += h[(size_t)row*D + c0 + j]*pv; pp += pv*pv; }
  for (int o=16;o>0;o>>=1){ dot += __shfl_xor(dot,o,32); pp += __shfl_xor(pp,o,32); }
  if (lane==0){
    float s = tanhf(dot * rsqrtf(pp));
    score[row] = s;
    unsigned u = __float_as_uint(s);
    keys[row] = (u & 0x80000000u) ? ~u : (u | 0x80000000u);
  }
}

// ---------------- radix select (k-th largest key) ----------------
__global__ void k_radix_init(SelState* st, unsigned* hist, int k){
  if (threadIdx.x==0){ st->prefix=0u; st->krem=(unsigned)k; st->total_gt=0u; }
  hist[threadIdx.x] = 0u;
}
__global__ void k_radix_pass(const unsigned* __restrict__ keys, int n,
                             const SelState* __restrict__ st, unsigned* hist, int shift){
  __shared__ unsigned lh[256];
  lh[threadIdx.x] = 0u; __syncthreads();
  unsigned prefix = st->prefix;
  unsigned maskhi = (shift==24) ? 0u : (0xFFFFFFFFu << (shift+8));
  int i = blockIdx.x*blockDim.x + threadIdx.x;
  if (i < n){
    unsigned kk = keys[i];
    if ((kk & maskhi) == prefix) atomicAdd(&lh[(kk>>shift)&255], 1u);
  }
  __syncthreads();
  if (lh[threadIdx.x]) atomicAdd(&hist[threadIdx.x], lh[threadIdx.x]);
}
__global__ void k_radix_scan(unsigned* hist, SelState* st, int shift){
  __shared__ unsigned h[256];
  h[threadIdx.x] = hist[threadIdx.x];
  __syncthreads();
  if (threadIdx.x==0){
    unsigned krem = st->krem, cum = 0u;
    int d = 255;
    for (; d >= 0; --d){ unsigned c = h[d]; if (cum + c >= krem) break; cum += c; }
    if (d < 0) d = 0;
    st->prefix |= ((unsigned)d) << shift;
    st->krem = krem - cum;
  }
  __syncthreads();
  hist[threadIdx.x] = 0u;
}

// ---------------- deterministic compaction of the top-k set ----------------
__global__ void k_topk_blockcount(const unsigned* __restrict__ keys, int n,
                                  const SelState* __restrict__ st, int* bsums){
  __shared__ int cg, ce;
  if (threadIdx.x==0){ cg=0; ce=0; } __syncthreads();
  int i = blockIdx.x*blockDim.x + threadIdx.x;
  if (i < n){
    unsigned kk = keys[i], T = st->prefix;
    if (kk > T) atomicAdd(&cg,1); else if (kk == T) atomicAdd(&ce,1);
  }
  __syncthreads();
  if (threadIdx.x==0){ bsums[2*blockIdx.x]=cg; bsums[2*blockIdx.x+1]=ce; }
}
__global__ void k_topk_scanblocks(const int* __restrict__ bsums, int* offs, int nb, SelState* st){
  if (threadIdx.x==0){
    int sg=0, se=0;
    for (int b=0;b<nb;++b){ offs[2*b]=sg; offs[2*b+1]=se; sg+=bsums[2*b]; se+=bsums[2*b+1]; }
    st->total_gt = (unsigned)sg;
  }
}
__global__ void k_topk_scatter(const unsigned* __restrict__ keys, int n,
                               const SelState* __restrict__ st, const int* __restrict__ offs,
                               int* remap, int* perm){
  __shared__ int sh[256];
  int b = blockIdx.x, tid = threadIdx.x, i = b*256 + tid;
  unsigned T = st->prefix, needed = st->krem, total_gt = st->total_gt;
  int gt=0, eq=0;
  if (i < n){ unsigned kk = keys[i]; gt = (kk > T); eq = (kk == T); }
  int val = gt | (eq << 16);
  sh[tid] = val; __syncthreads();
  for (int o=1;o<256;o<<=1){
    int t = (tid>=o) ? sh[tid-o] : 0;
    __syncthreads();
    sh[tid] += t;
    __syncthreads();
  }
  int excl = sh[tid] - val;
  int rank_gt = excl & 0xFFFF, rank_eq = excl >> 16;
  if (i < n){
    int pos = -1;
    if (gt) pos = offs[2*b] + rank_gt;
    else if (eq){
      int ge = offs[2*b+1] + rank_eq;
      if ((unsigned)ge < needed) pos = (int)total_gt + ge;
    }
    remap[i] = pos;
    if (pos >= 0) perm[pos] = i;
  }
}
__global__ void k_gather(const float* __restrict__ h, const float* __restrict__ score,
                         const int* __restrict__ perm, float* __restrict__ out, int k){
  int i = blockIdx.x*blockDim.x + threadIdx.x;
  if (i < k*D){
    int pos = i>>7, c = i&127;
    int old = perm[pos];
    out[i] = h[(size_t)old*D + c] * score[old];
  }
}
__global__ void k_edge_remap(const int* __restrict__ remap, int* es, int* ed, int* em, int E){
  int e = blockIdx.x*blockDim.x + threadIdx.x;
  if (e < E){
    if (em[e]){
      int ns = remap[es[e]], nd = remap[ed[e]];
      if (ns < 0 || nd < 0){ em[e]=0; es[e]=0; ed[e]=0; }
      else { es[e]=ns; ed[e]=nd; }
    } else { es[e]=0; ed[e]=0; }
  }
}

// ---------------- global max + mean pool ----------------
__global__ void k_pool_partial(const float* __restrict__ h, int k, float* part){
  int c = threadIdx.x;                                   // 128 columns
  float mx = -3.402823466e38f, sm = 0.f;
  for (int r = blockIdx.x; r < k; r += PB){
    float v = h[(size_t)r*D + c];
    mx = fmaxf(mx, v); sm += v;
  }
  part[blockIdx.x*256 + c]       = mx;
  part[blockIdx.x*256 + 128 + c] = sm;
}
__global__ void k_pool_final(const float* __restrict__ part, float* reads, int layer, float invk){
  int c = threadIdx.x;
  float mx = -3.402823466e38f, sm = 0.f;
  for (int b=0;b<PB;++b){ mx = fmaxf(mx, part[b*256+c]); sm += part[b*256+128+c]; }
  reads[layer*256 + c]       = mx;
  reads[layer*256 + 128 + c] = sm * invk;
}

// ---------------- MLP head ----------------
__global__ void k_lin1(const float* __restrict__ r, const float* __restrict__ w,
                       const float* __restrict__ b, const float* __restrict__ pa, float* z){
  int j = blockIdx.x*blockDim.x + threadIdx.x;
  if (j < 1280){
    float s = b[j];
    for (int i=0;i<2560;++i) s += r[i]*w[(size_t)i*1280 + j];
    float a = pa[0];
    z[j] = s > 0.f ? s : a*s;
  }
}
__global__ void k_lin2_norm(const float* __restrict__ z1, const float* __restrict__ w,
                            const float* __restrict__ b, const float* __restrict__ pa,
                            float* out){
  __shared__ float zz[8];
  int wid = threadIdx.x>>5, lane = threadIdx.x&31;
  float s = 0.f;
  for (int i=lane;i<1280;i+=32) s += z1[i]*w[(size_t)i*8 + wid];
  for (int o=16;o>0;o>>=1) s += __shfl_xor(s,o,32);
  if (lane==0) zz[wid] = s + b[wid];
  __syncthreads();
  if (threadIdx.x==0){
    float a = pa[0];
    float z[8], mn = 3.402823466e38f, mx = -3.402823466e38f, sum = 0.f;
    for (int o=0;o<8;++o){ float v = zz[o]; v = v>0.f ? v : a*v; z[o]=v; mn = fminf(mn,v); }
    for (int o=0;o<8;++o){ z[o] -= mn; mx = fmaxf(mx, z[o]); }
    for (int o=0;o<8;++o){ z[o] /= mx; sum += z[o]; }
    for (int o=0;o<8;++o) out[o] = z[o]/sum;
  }
}

extern "C" void kernel_launch(void* const* d_in, const int* in_sizes, int n_in,
                              void* d_out, int out_size, void* d_ws, size_t ws_size,
                              hipStream_t stream){
  (void)n_in; (void)out_size; (void)ws_size;
  const float* x      = (const float*)d_in[0];
  const int*   ei     = (const int*)  d_in[1];
  const float* W1     = (const float*)d_in[2];
  const float* V1     = (const float*)d_in[3];
  const float* Ws     = (const float*)d_in[4];
  const float* Vs     = (const float*)d_in[5];
  const float* conv_b = (const float*)d_in[6];
  const float* bng    = (const float*)d_in[7];
  const float* bnb    = (const float*)d_in[8];
  const float* bnm    = (const float*)d_in[9];
  const float* bnv    = (const float*)d_in[10];
  const float* poolp  = (const float*)d_in[11];
  const float* pa     = (const float*)d_in[12];
  const float* l1w    = (const float*)d_in[13];
  const float* l1b    = (const float*)d_in[14];
  const float* l2w    = (const float*)d_in[15];
  const float* l2b    = (const float*)d_in[16];

  const int N0 = in_sizes[0];
  const int E  = in_sizes[1]/2;

  char* base = (char*)d_ws;
  size_t off = 0;
  auto carve = [&](size_t bytes)->char*{
    char* p = base + off; off = (off + bytes + 255) & ~(size_t)255; return p; };

  float*    hA    = (float*)   carve((size_t)N0*D*4);
  float*    hB    = (float*)   carve((size_t)N0*D*4);
  float*    hw    = (float*)   carve((size_t)N0*D*4);
  float*    acc   = (float*)   carve((size_t)N0*D*4);
  float*    deg   = (float*)   carve((size_t)N0*4);
  float*    dinv  = (float*)   carve((size_t)N0*4);
  float*    score = (float*)   carve((size_t)N0*4);
  unsigned* keys  = (unsigned*)carve((size_t)N0*4);
  int*      remap = (int*)     carve((size_t)N0*4);
  int*      perm  = (int*)     carve((size_t)N0*4);
  int*      es    = (int*)     carve((size_t)E*4);
  int*      ed    = (int*)     carve((size_t)E*4);
  int*      em    = (int*)     carve((size_t)E*4);
  unsigned* hist  = (unsigned*)carve(256*4);
  SelState* st    = (SelState*)carve(sizeof(SelState));
  int nbmax = (N0+255)/256;
  int*      bsums = (int*)     carve((size_t)2*nbmax*4);
  int*      offs  = (int*)     carve((size_t)2*nbmax*4);
  float*    reads = (float*)   carve(2560*4);
  float*    part  = (float*)   carve((size_t)PB*256*4);
  float*    z1    = (float*)   carve(1280*4);

  auto cdiv = [](int a, int b){ return (a + b - 1)/b; };

  k_edges_init<<<cdiv(E,256),256,0,stream>>>(ei, es, ed, em, E);

  float* hcur = hA; float* hnext = hB;
  int n = N0;
  for (int L=0; L<10; ++L){
    if (L==0){
      k_layer0<<<cdiv(n*D,256),256,0,stream>>>(x, W1, V1, conv_b, hw, acc, n);
    } else {
      const float* W = Ws + (size_t)(L-1)*D*D;
      const float* V = Vs + (size_t)(L-1)*D*D;
      k_gemm<<<cdiv(n,64),256,0,stream>>>(hcur, W, nullptr, hw, n);
      k_gemm<<<cdiv(n,64),256,0,stream>>>(hcur, V, conv_b + (size_t)L*D, acc, n);
    }
    k_zero<<<cdiv(n,256),256,0,stream>>>(deg, n);
    k_edge_deg<<<cdiv(E,256),256,0,stream>>>(ed, em, deg, E);
    k_dinv<<<cdiv(n,256),256,0,stream>>>(deg, dinv, n);
    k_edge_agg<<<cdiv(E*32,256),256,0,stream>>>(es, ed, em, dinv, hw, acc, E);
    k_post<<<cdiv(n*D,256),256,0,stream>>>(acc, bng+(size_t)L*D, bnb+(size_t)L*D,
                                           bnm+(size_t)L*D, bnv+(size_t)L*D, pa, hw, n);
    k_score<<<cdiv(n,8),256,0,stream>>>(hw, poolp+(size_t)L*D, score, keys, n);

    int k = (4*n + 4)/5;               // ceil(0.8 n)
    k_radix_init<<<1,256,0,stream>>>(st, hist, k);
    for (int p=0;p<4;++p){
      k_radix_pass<<<cdiv(n,256),256,0,stream>>>(keys, n, st, hist, 24-8*p);
      k_radix_scan<<<1,256,0,stream>>>(hist, st, 24-8*p);
    }
    int nb = cdiv(n,256);
    k_topk_blockcount<<<nb,256,0,stream>>>(keys, n, st, bsums);
    k_topk_scanblocks<<<1,64,0,stream>>>(bsums, offs, nb, st);
    k_topk_scatter<<<nb,256,0,stream>>>(keys, n, st, offs, remap, perm);
    k_gather<<<cdiv(k*D,256),256,0,stream>>>(hw, score, perm, hnext, k);
    k_edge_remap<<<cdiv(E,256),256,0,stream>>>(remap, es, ed, em, E);
    k_pool_partial<<<PB,128,0,stream>>>(hnext, k, part);
    k_pool_final<<<1,128,0,stream>>>(part, reads, L, 1.0f/(float)k);

    n = k;
    float* t = hcur; hcur = hnext; hnext = t;
  }
  k_lin1<<<cdiv(1280,256),256,0,stream>>>(reads, l1w, l1b, pa, z1);
  k_lin2_norm<<<1,256,0,stream>>>(z1, l2w, l2b, pa, (float*)d_out);
}